// VectorQuantizer_17660905521540
// MI455X (gfx1250) — compile-verified
//
#include <hip/hip_runtime.h>
#include <hip/hip_bf16.h>
#include <stdint.h>

typedef float v2f __attribute__((ext_vector_type(2)));
typedef float v8f __attribute__((ext_vector_type(8)));
typedef unsigned u32x4 __attribute__((ext_vector_type(4)));
typedef unsigned u32x8 __attribute__((ext_vector_type(8)));

#define N_TOT   16384   // 16*32*32
#define D_DIM   256
#define K_TOT   4096
#define BM      128     // rows per block
#define CHUNK   64      // codebook columns per iteration
#define KSPLIT  2       // codebook halves processed by different blocks
#define KHALF   (K_TOT / KSPLIT)
#define STR     260     // padded LDS row stride (multiple of 4, 16B rows, bank-skewed)

// Order-preserving float<->uint mapping so atomicMin(uint) == float min.
__device__ __forceinline__ unsigned f2mono(float f) {
    unsigned u = __float_as_uint(f);
    return u ^ ((unsigned)((int)u >> 31) | 0x80000000u);
}
__device__ __forceinline__ float mono2f(unsigned u) {
    unsigned b = (u & 0x80000000u) ? (u ^ 0x80000000u) : ~u;
    return __uint_as_float(b);
}

// Issue one TDM DMA: codebook tile [64 rows x 256 f32] at row `kb` -> LDS buffer,
// padded +4 DWORDs per 256-DWORD row (=> row stride 260 f32, bank-skewed).
__device__ __forceinline__ void tdm_load_chunk(const float* cb, int kb, unsigned ldsOff) {
    unsigned long long ga = (unsigned long long)(uintptr_t)cb
                          + (unsigned long long)kb * (D_DIM * 4ull);
    u32x4 g0;
    g0.x = 1u;                                   // count=1 (valid user descriptor)
    g0.y = ldsOff;                               // lds_addr (bytes)
    g0.z = (unsigned)ga;                         // global_addr[31:0]
    g0.w = (unsigned)((ga >> 32) & 0x01FFFFFFull) | 0x80000000u; // addr[56:32] | type=2

    u32x8 g1;
    g1.s0 = (2u << 16)      // data_size = 4 bytes
          | (1u << 20)      // pad_enable
          | (7u << 22)      // pad_interval: 256 DWORDs
          | (3u << 25);     // pad_amount: 4 DWORDs
    g1.s1 = ((unsigned)D_DIM & 0xFFFFu) << 16;               // tensor_dim0[15:0]
    g1.s2 = ((unsigned)K_TOT & 0xFFFFu) << 16;               // dim0 hi=0 | tensor_dim1[15:0]
    g1.s3 = ((unsigned)D_DIM & 0xFFFFu) << 16;               // dim1 hi=0 | tile_dim0=256
    g1.s4 = (unsigned)CHUNK;                                 // tile_dim1=64 | tile_dim2=0
    g1.s5 = (unsigned)D_DIM;                                 // tensor_dim0_stride[31:0]
    g1.s6 = 0u;                                              // stride hi | dim1_stride lo
    g1.s7 = 0u;
    asm volatile("tensor_load_to_lds %0, %1" :: "s"(g0), "s"(g1) : "memory");
}

__global__ __launch_bounds__(256)
void vq_init_kernel(unsigned long long* __restrict__ rowPack,
                    unsigned* __restrict__ minBits, int* __restrict__ active) {
    int i = blockIdx.x * 256 + threadIdx.x;
    if (i < N_TOT) rowPack[i] = 0xFFFFFFFFFFFFFFFFull;
    if (i < K_TOT) {
        minBits[i] = 0xFFFFFFFFu;   // +inf in monotone space
        active[i]  = 0;
    }
}

// Fused distance GEMM (fp32 WMMA, TDM-fed double-buffered LDS) + row argmin + column min.
// Grid: (N/BM) row-tiles x KSPLIT codebook halves.
__global__ __launch_bounds__(256)
void vq_dist_kernel(const float* __restrict__ x, const float* __restrict__ cb,
                    unsigned long long* __restrict__ rowPack,
                    unsigned* __restrict__ minBits) {
    __shared__ float lds_x[BM * STR];
    __shared__ float lds_c[2][CHUNK * STR];
    __shared__ float xsq[BM];
    __shared__ float csq[CHUNK];

    const int tid   = threadIdx.x;
    const int lane  = tid & 31;
    const int wave  = tid >> 5;        // 0..7 (all stacked in M)
    const int li    = lane & 15;
    const int hh    = lane >> 4;       // half-wave selector
    const int m0    = wave * 16;
    const int rowBase = (blockIdx.x >> 1) * BM;
    const int kBase   = (blockIdx.x & 1) * KHALF;

    const unsigned ldsOffC0 = (unsigned)(uintptr_t)(&lds_c[0][0]);
    const unsigned ldsOffC1 = (unsigned)(uintptr_t)(&lds_c[1][0]);

    // Prefetch first chunk via TDM while we stage the x tile.
    if (wave == 0) tdm_load_chunk(cb, kBase, ldsOffC0);

    // ---- load x tile (2 threads per row, 32 float4 each) ----
    {
        const int r    = tid >> 1;
        const int cseg = (tid & 1) * 128;
        const float4* src = (const float4*)(x + (size_t)(rowBase + r) * D_DIM + cseg);
        float4*       dst = (float4*)(lds_x + r * STR + cseg);
        #pragma unroll
        for (int i = 0; i < 32; ++i) dst[i] = src[i];
    }
    __syncthreads();
    if (tid < BM) {                    // deterministic row ||x||^2
        float s = 0.f;
        const float* row = lds_x + tid * STR;
        for (int d = 0; d < D_DIM; ++d) s = fmaf(row[d], row[d], s);
        xsq[tid] = s;
    }
    __syncthreads();

    float xs[8];
    #pragma unroll
    for (int r = 0; r < 8; ++r) xs[r] = xsq[m0 + 8 * hh + r];

    float rmin[8]; int rk[8];
    #pragma unroll
    for (int r = 0; r < 8; ++r) { rmin[r] = 3.4e38f; rk[r] = 0; }

    for (int c = 0; c < KHALF / CHUNK; ++c) {
        const int kb  = kBase + c * CHUNK;
        const int cur = c & 1;
        const float* ldsC = &lds_c[cur][0];

        if (wave == 0) __builtin_amdgcn_s_wait_tensorcnt(0); // chunk c landed in buf[cur]
        __syncthreads();            // buf[cur] ready; buf[cur^1] compute retired
        if (wave == 0 && c + 1 < KHALF / CHUNK)
            tdm_load_chunk(cb, kb + CHUNK, cur ? ldsOffC0 : ldsOffC1); // overlap next DMA

        if (tid < CHUNK) {          // deterministic row ||c||^2
            float s = 0.f;
            const float* row = ldsC + tid * STR;
            for (int d = 0; d < D_DIM; ++d) s = fmaf(row[d], row[d], s);
            csq[tid] = s;
        }
        __syncthreads();

        // ---- 16x64 cross-product strip via fp32 WMMA: 4 independent chains ----
        v8f acc0 = {}; v8f acc1 = {}; v8f acc2 = {}; v8f acc3 = {};
        const float* ax  = lds_x + (m0 + li) * STR + 2 * hh;
        const float* bp  = ldsC + li * STR + 2 * hh;
        #pragma unroll 8
        for (int d = 0; d < D_DIM; d += 4) {
            v2f a  = *(const v2f*)(ax + d);
            v2f b0 = *(const v2f*)(bp + d);
            v2f b1 = *(const v2f*)(bp + 16 * STR + d);
            v2f b2 = *(const v2f*)(bp + 32 * STR + d);
            v2f b3 = *(const v2f*)(bp + 48 * STR + d);
            acc0 = __builtin_amdgcn_wmma_f32_16x16x4_f32(false, a, false, b0,
                                                         (short)0, acc0, false, false);
            acc1 = __builtin_amdgcn_wmma_f32_16x16x4_f32(false, a, false, b1,
                                                         (short)0, acc1, false, false);
            acc2 = __builtin_amdgcn_wmma_f32_16x16x4_f32(false, a, false, b2,
                                                         (short)0, acc2, false, false);
            acc3 = __builtin_amdgcn_wmma_f32_16x16x4_f32(false, a, false, b3,
                                                         (short)0, acc3, false, false);
        }

        // ---- distances + running row-argmin + column-min ----
        const float cs0 = csq[li];
        const float cs1 = csq[16 + li];
        const float cs2 = csq[32 + li];
        const float cs3 = csq[48 + li];
        const int   k0  = kb + li;
        float cmin0 = 3.4e38f, cmin1 = 3.4e38f, cmin2 = 3.4e38f, cmin3 = 3.4e38f;
        #pragma unroll
        for (int r = 0; r < 8; ++r) {
            float d0 = fmaf(-2.f, acc0[r], xs[r] + cs0);
            float d1 = fmaf(-2.f, acc1[r], xs[r] + cs1);
            float d2 = fmaf(-2.f, acc2[r], xs[r] + cs2);
            float d3 = fmaf(-2.f, acc3[r], xs[r] + cs3);
            if (d0 < rmin[r]) { rmin[r] = d0; rk[r] = k0; }
            if (d1 < rmin[r]) { rmin[r] = d1; rk[r] = k0 + 16; }
            if (d2 < rmin[r]) { rmin[r] = d2; rk[r] = k0 + 32; }
            if (d3 < rmin[r]) { rmin[r] = d3; rk[r] = k0 + 48; }
            cmin0 = fminf(cmin0, d0);
            cmin1 = fminf(cmin1, d1);
            cmin2 = fminf(cmin2, d2);
            cmin3 = fminf(cmin3, d3);
        }
        cmin0 = fminf(cmin0, __shfl_xor(cmin0, 16, 32));
        cmin1 = fminf(cmin1, __shfl_xor(cmin1, 16, 32));
        cmin2 = fminf(cmin2, __shfl_xor(cmin2, 16, 32));
        cmin3 = fminf(cmin3, __shfl_xor(cmin3, 16, 32));
        if (hh == 0) {
            atomicMin(&minBits[k0],      f2mono(cmin0));
            atomicMin(&minBits[k0 + 16], f2mono(cmin1));
            atomicMin(&minBits[k0 + 32], f2mono(cmin2));
            atomicMin(&minBits[k0 + 48], f2mono(cmin3));
        }
    }

    // ---- final row argmin: butterfly inside each half-wave, then packed u64 atomicMin.
    // pack = (monotone_dist << 32) | k : min(pack) == (min dist, first index on tie).
    #pragma unroll
    for (int r = 0; r < 8; ++r) {
        float rv = rmin[r]; int ri = rk[r];
        #pragma unroll
        for (int off = 1; off < 16; off <<= 1) {
            float ov = __shfl_xor(rv, off, 32);
            int   oi = __shfl_xor(ri, off, 32);
            if (ov < rv || (ov == rv && oi < ri)) { rv = ov; ri = oi; }
        }
        if (li == 0) {
            unsigned long long pack =
                ((unsigned long long)f2mono(rv) << 32) | (unsigned)ri;
            atomicMin(&rowPack[rowBase + m0 + 8 * hh + r], pack);
        }
    }
}

// Gather codebook by token -> out; per-block MSE partials (deterministic); active flags.
__global__ __launch_bounds__(256)
void vq_gather_kernel(const float* __restrict__ x, const float* __restrict__ cb,
                      const unsigned long long* __restrict__ rowPack,
                      float* __restrict__ out,
                      float* __restrict__ blockSums, int* __restrict__ active) {
    __shared__ float red[256];
    const int n   = blockIdx.x;
    const int tid = threadIdx.x;
    const int t   = (int)(unsigned)(rowPack[n] & 0xFFFFFFFFull);
    const float c  = cb[(size_t)t * D_DIM + tid];
    const float xv = x[(size_t)n * D_DIM + tid];
    out[(size_t)n * D_DIM + tid] = c;       // straight-through == embeddings
    const float diff = c - xv;
    red[tid] = diff * diff;
    if (tid == 0) active[t] = 1;            // idempotent, race-free by value
    __syncthreads();
    for (int s = 128; s > 0; s >>= 1) {
        if (tid < s) red[tid] += red[tid + s];
        __syncthreads();
    }
    if (tid == 0) blockSums[n] = red[0];
}

__global__ __launch_bounds__(256)
void vq_finalize_kernel(const float* __restrict__ blockSums,
                        const unsigned* __restrict__ minBits,
                        const int* __restrict__ active,
                        float* __restrict__ out) {
    __shared__ float red[256];
    const int tid = threadIdx.x;

    float mse = 0.f;
    for (int i = tid; i < N_TOT; i += 256) mse += blockSums[i];
    red[tid] = mse; __syncthreads();
    for (int s = 128; s > 0; s >>= 1) { if (tid < s) red[tid] += red[tid + s]; __syncthreads(); }
    const float mseTot = red[0];
    __syncthreads();

    float ent = 0.f;
    for (int k = tid; k < K_TOT; k += 256)
        if (active[k] == 0) ent += mono2f(minBits[k]);
    red[tid] = ent; __syncthreads();
    for (int s = 128; s > 0; s >>= 1) { if (tid < s) red[tid] += red[tid + s]; __syncthreads(); }

    if (tid == 0) {
        // embedding (1.0) + commitment (0.25) losses share the same value numerically
        float loss = 1.25f * (mseTot / (float)(N_TOT * D_DIM))
                   + 0.02f * (red[0] / (float)K_TOT);
        out[(size_t)N_TOT * D_DIM] = loss;
    }
}

extern "C" void kernel_launch(void* const* d_in, const int* in_sizes, int n_in,
                              void* d_out, int out_size, void* d_ws, size_t ws_size,
                              hipStream_t stream) {
    const float* x  = (const float*)d_in[0];   // [16,32,32,256]
    const float* cb = (const float*)d_in[1];   // [4096,256]
    float* out = (float*)d_out;                // 4194304 floats + 1 loss

    char* ws = (char*)d_ws;
    unsigned long long* rowPack = (unsigned long long*)ws;        // 16384 * 8 B
    unsigned* minBits   = (unsigned*)(ws + 131072);               //  4096 * 4 B
    int*      active    = (int*)(ws + 147456);                    //  4096 * 4 B
    float*    blockSums = (float*)(ws + 163840);                  // 16384 * 4 B

    vq_init_kernel<<<N_TOT / 256, 256, 0, stream>>>(rowPack, minBits, active);
    vq_dist_kernel<<<(N_TOT / BM) * KSPLIT, 256, 0, stream>>>(x, cb, rowPack, minBits);
    vq_gather_kernel<<<N_TOT, 256, 0, stream>>>(x, cb, rowPack, out, blockSums, active);
    vq_finalize_kernel<<<1, 256, 0, stream>>>(blockSums, minBits, active, out);
}